// SAGE_12936441496236
// MI455X (gfx1250) — compile-verified
//
#include <hip/hip_runtime.h>
#include <hip/hip_bf16.h>

typedef __attribute__((ext_vector_type(2))) float v2f;
typedef __attribute__((ext_vector_type(8))) float v8f;

#define IN_F   128
#define HID    128
#define NCLS   64
#define MTILE  32
#define LDSPAD 132   // 128 + 4 -> rows land on rotating banks (stride 4 mod 64)

// Guaranteed hardware FP32 atomic add (no CAS fallback).
__device__ __forceinline__ void atomic_add_f32(float* addr, float val) {
    asm volatile("global_atomic_add_f32 %0, %1, off"
                 :: "v"(addr), "v"(val) : "memory");
}

// agg := x  (the +x[v] self term), deg := 1.0f (the +1 term)
__global__ void __launch_bounds__(256) sage_init_kernel(
    const float* __restrict__ x, float* __restrict__ agg,
    float* __restrict__ deg, int n_nodes)
{
    int i = blockIdx.x * 256 + threadIdx.x;          // float4 index
    int n4 = n_nodes * (IN_F / 4);
    if (i < n4) {
        ((float4*)agg)[i] = ((const float4*)x)[i];
    }
    if (i < n_nodes) deg[i] = 1.0f;
}

// One wave32 per edge: lane l accumulates features [4l, 4l+4)
__global__ void __launch_bounds__(256) sage_edge_kernel(
    const float* __restrict__ x, const int* __restrict__ src,
    const int* __restrict__ dst, float* __restrict__ agg,
    float* __restrict__ deg, int n_edges)
{
    int wid  = (blockIdx.x * 256 + threadIdx.x) >> 5;
    int lane = threadIdx.x & 31;
    if (wid >= n_edges) return;
    int s = src[wid];
    int d = dst[wid];
    const float4 xv = ((const float4*)(x + (size_t)s * IN_F))[lane];
    float* ap = agg + (size_t)d * IN_F + lane * 4;
    atomic_add_f32(ap + 0, xv.x);
    atomic_add_f32(ap + 1, xv.y);
    atomic_add_f32(ap + 2, xv.z);
    atomic_add_f32(ap + 3, xv.w);
    if (lane == 0) atomic_add_f32(deg + d, 1.0f);
}

// Fused: normalize -> GEMM1(+bias,ReLU) -> emb -> GEMM2(+bias) -> out
// Block = 256 threads = 8 waves, 32-row M tile. All global output stores are
// coalesced float4 streams from LDS staging tiles.
__global__ void __launch_bounds__(256) sage_gemm_kernel(
    const float* __restrict__ agg, const float* __restrict__ deg,
    const float* __restrict__ W1, const float* __restrict__ b1,
    const float* __restrict__ W2, const float* __restrict__ b2,
    float* __restrict__ out, float* __restrict__ emb, int n_nodes)
{
    __shared__ __align__(16) float As[MTILE][LDSPAD]; // h_neigh tile, later out tile
    __shared__ __align__(16) float Es[MTILE][LDSPAD]; // relu(emb) tile
    __shared__ float Ds[MTILE];                       // 1/(deg+1)

    const int tid  = threadIdx.x;
    const int row0 = blockIdx.x * MTILE;

    if (tid < MTILE) {
        int r = row0 + tid;
        Ds[tid] = (r < n_nodes) ? (1.0f / deg[r]) : 0.0f;
    }
    __syncthreads();

    for (int i = tid; i < MTILE * IN_F; i += 256) {
        int r = i >> 7, k = i & (IN_F - 1);
        int gr = row0 + r;
        As[r][k] = (gr < n_nodes) ? agg[(size_t)gr * IN_F + k] * Ds[r] : 0.0f;
    }
    __syncthreads();

    const int lane = tid & 31;
    const int wave = tid >> 5;        // 0..7
    const int half = lane >> 4;       // 0: K+0/1 rows, 1: K+2/3 rows (A/B layout)
    const int l16  = lane & 15;

    // ---- GEMM1: Es = relu(As @ W1 + b1); wave = N-tile, 2 M-subtiles ----
    {
        const int nb = wave * 16 + l16;          // column 0..127
        const float bias = b1[nb];
        v8f acc0, acc1;
        #pragma unroll
        for (int v = 0; v < 8; ++v) { acc0[v] = bias; acc1[v] = bias; }

        #pragma unroll 8
        for (int ks = 0; ks < IN_F / 4; ++ks) {
            const int k = ks * 4 + 2 * half;
            v2f a0, a1, b;
            a0.x = As[l16][k];        a0.y = As[l16][k + 1];
            a1.x = As[16 + l16][k];   a1.y = As[16 + l16][k + 1];
            b.x  = W1[k * HID + nb];  b.y  = W1[(k + 1) * HID + nb];
            acc0 = __builtin_amdgcn_wmma_f32_16x16x4_f32(
                       false, a0, false, b, (short)0, acc0, false, false);
            acc1 = __builtin_amdgcn_wmma_f32_16x16x4_f32(
                       false, a1, false, b, (short)0, acc1, false, false);
        }

        #pragma unroll
        for (int v = 0; v < 8; ++v) {
            const int m = v + 8 * half;          // C/D layout: M = vgpr + 8*half
            Es[m][nb]      = acc0[v] > 0.0f ? acc0[v] : 0.0f;
            Es[16 + m][nb] = acc1[v] > 0.0f ? acc1[v] : 0.0f;
        }
    }
    __syncthreads();   // Es complete; As no longer read

    // ---- GEMM2: As(reused) = Es @ W2 + b2; wave -> (msub, ntile) ----
    {
        const int msub = wave >> 2;              // 0..1
        const int nt   = wave & 3;               // 0..3
        const int n2   = nt * 16 + l16;          // column 0..63
        const float bias = b2[n2];
        v8f acc;
        #pragma unroll
        for (int v = 0; v < 8; ++v) acc[v] = bias;

        #pragma unroll 8
        for (int ks = 0; ks < HID / 4; ++ks) {
            const int k = ks * 4 + 2 * half;
            v2f a, b;
            a.x = Es[msub * 16 + l16][k];  a.y = Es[msub * 16 + l16][k + 1];
            b.x = W2[k * NCLS + n2];       b.y = W2[(k + 1) * NCLS + n2];
            acc = __builtin_amdgcn_wmma_f32_16x16x4_f32(
                      false, a, false, b, (short)0, acc, false, false);
        }

        #pragma unroll
        for (int v = 0; v < 8; ++v) {
            As[msub * 16 + v + 8 * half][n2] = acc[v];   // stage out tile in LDS
        }
    }
    __syncthreads();

    // ---- Coalesced float4 stores of both outputs from LDS staging ----
    #pragma unroll
    for (int i = tid; i < MTILE * (HID / 4); i += 256) {   // 1024 float4, 4 iters
        int r = i >> 5, c = i & 31;
        int g = row0 + r;
        if (g < n_nodes)
            ((float4*)emb)[(size_t)g * (HID / 4) + c] = *(const float4*)&Es[r][c * 4];
    }
    #pragma unroll
    for (int i = tid; i < MTILE * (NCLS / 4); i += 256) {  // 512 float4, 2 iters
        int r = i >> 4, c = i & 15;
        int g = row0 + r;
        if (g < n_nodes)
            ((float4*)out)[(size_t)g * (NCLS / 4) + c] = *(const float4*)&As[r][c * 4];
    }
}

extern "C" void kernel_launch(void* const* d_in, const int* in_sizes, int n_in,
                              void* d_out, int out_size, void* d_ws, size_t ws_size,
                              hipStream_t stream) {
    const float* x   = (const float*)d_in[0];
    const int*   src = (const int*)d_in[1];
    const int*   dst = (const int*)d_in[2];
    const float* W1  = (const float*)d_in[3];
    const float* b1  = (const float*)d_in[4];
    const float* W2  = (const float*)d_in[5];
    const float* b2  = (const float*)d_in[6];

    const int n_nodes = in_sizes[0] / IN_F;
    const int n_edges = in_sizes[1];

    float* agg = (float*)d_ws;                          // n_nodes*128 floats
    float* deg = agg + (size_t)n_nodes * IN_F;          // n_nodes floats

    float* out = (float*)d_out;                         // n_nodes*64
    float* emb = out + (size_t)n_nodes * NCLS;          // n_nodes*128

    {   // agg := x, deg := 1
        int n4 = n_nodes * (IN_F / 4);
        sage_init_kernel<<<(n4 + 255) / 256, 256, 0, stream>>>(x, agg, deg, n_nodes);
    }
    {   // edge scatter: one wave per edge
        long long threads = (long long)n_edges * 32;
        int blocks = (int)((threads + 255) / 256);
        sage_edge_kernel<<<blocks, 256, 0, stream>>>(x, src, dst, agg, deg, n_edges);
    }
    {   // fused normalize + GEMM1 + ReLU + GEMM2
        int blocks = (n_nodes + MTILE - 1) / MTILE;
        sage_gemm_kernel<<<blocks, 256, 0, stream>>>(agg, deg, W1, b1, W2, b2,
                                                     out, emb, n_nodes);
    }
}